// sisoLoCoPicDetector_64811056497379
// MI455X (gfx1250) — compile-verified
//
#include <hip/hip_runtime.h>
#include <math.h>

typedef float v2f __attribute__((ext_vector_type(2)));
typedef float v8f __attribute__((ext_vector_type(8)));

#define EPSF  1e-4f
#define EPS2F 1e-8f
#define QAMN  0.31622776601683794f  // 1/sqrt(10)

// 16-QAM constellation (binary_repr MSB-first per reference)
__device__ __host__ constexpr float ptsr(int p) {
  int a0 = (p >> 3) & 1, a2 = (p >> 1) & 1;
  return QAMN * (float)((1 - 2 * a0) * (2 - (1 - 2 * a2)));
}
__device__ __host__ constexpr float ptsi(int p) {
  int a1 = (p >> 2) & 1, a3 = p & 1;
  return QAMN * (float)((1 - 2 * a1) * (2 - (1 - 2 * a3)));
}

__global__ __launch_bounds__(256)
void siso_mmse_pic_kernel(const float* __restrict__ yp,
                          const float* __restrict__ hp,
                          const float* __restrict__ llrp,
                          const float* __restrict__ nop,
                          float* __restrict__ outp,
                          int nsub) {
  // Per-wave 32x16 staging matrix W (rows = [Hr;Hi], cols: U(0-3), V=[-Hi;Hr](4-7),
  // [Yr;Yi](8), [-Yi;Yr](9), zero-pad 10-15). W^T W gives ReG / -ImG / Re ymf / -Im ymf.
  __shared__ float Wsh[8][32 * 16];
  __shared__ float Gbuf[256][41];  // 40 used; stride 41 -> no bank conflicts

  const int tid = threadIdx.x;
  const int wv  = tid >> 5;
  const int ln  = tid & 31;
  const int blockBase = blockIdx.x * 256;

  float* Wf = Wsh[wv];

  // zero-pad columns 10..15 of this lane's row (once; fenced by first dscnt wait)
#pragma unroll
  for (int c = 10; c < 16; ++c) Wf[ln * 16 + c] = 0.f;

  const int m  = ln >> 1;
  const int k0 = (ln & 1) * 2;

  // ---------------- Phase 1: Gram + matched filter via WMMA, one problem per wave-iter
  for (int j = 0; j < 32; ++j) {
    const int s = blockBase + wv * 32 + j;
    if (s < nsub) {
      // h: 128 contiguous floats per subcarrier (m-major, (k,re/im) inner) -> coalesced float4
      const float4 hv = ((const float4*)(hp + (size_t)s * 128))[ln];
      Wf[m * 16 + k0]              = hv.x;   // Re h[m][k0]
      Wf[m * 16 + 4 + k0]          = -hv.y;  // -Im
      Wf[(16 + m) * 16 + k0]       = hv.y;   // Im
      Wf[(16 + m) * 16 + 4 + k0]   = hv.x;   // Re
      Wf[m * 16 + k0 + 1]          = hv.z;
      Wf[m * 16 + 5 + k0]          = -hv.w;
      Wf[(16 + m) * 16 + k0 + 1]   = hv.w;
      Wf[(16 + m) * 16 + 5 + k0]   = hv.z;

      const float yv = yp[(size_t)s * 32 + ln];
      if ((ln & 1) == 0) { Wf[m * 16 + 8] = yv;  Wf[(16 + m) * 16 + 9] = yv; }
      else               { Wf[(16 + m) * 16 + 8] = yv;  Wf[m * 16 + 9] = -yv; }

      // DS is in-order within a wave; fence keeps compiler from reordering + drains stores
      asm volatile("s_wait_dscnt 0x0" ::: "memory");

      // D = W^T W via 8x V_WMMA_F32_16X16X4_F32 (A == B: Gram). f32 A 16x4 layout:
      // lanes 0-15 hold K=0 (vgpr0) / K=1 (vgpr1); lanes 16-31 hold K=2 / K=3.
      v8f acc = {0.f, 0.f, 0.f, 0.f, 0.f, 0.f, 0.f, 0.f};
      const int colI = ln & 15;
      const int sel  = (ln >= 16) ? 2 : 0;
#pragma unroll
      for (int c = 0; c < 8; ++c) {
        v2f av;
        av.x = Wf[(4 * c + sel) * 16 + colI];
        av.y = Wf[(4 * c + sel + 1) * 16 + colI];
        acc = __builtin_amdgcn_wmma_f32_16x16x4_f32(
            /*neg_a=*/false, av, /*neg_b=*/false, av,
            /*c_mod=*/(short)0, acc, /*reuse_a=*/false, /*reuse_b=*/false);
      }
      // D rows 0-3 (lanes 0-15 hold M=0..7 in acc[0..7], N=lane); keep cols 0..9
      if (ln < 10) {
        float* g = Gbuf[wv * 32 + j];
        g[ln]      = acc[0];
        g[10 + ln] = acc[1];
        g[20 + ln] = acc[2];
        g[30 + ln] = acc[3];
      }
      asm volatile("s_wait_dscnt 0x0" ::: "memory");  // WAR: before next j rewrites Wf
    }
  }

  __syncthreads();

  // ---------------- Phase 2: one thread per subcarrier, all-register serial math
  const int s2 = blockBase + tid;
  if (s2 >= nsub) return;

  const float* gb = Gbuf[tid];  // thread t's problem == the one its wave staged at j=lane
#define RG(i, jj) (gb[(i) * 10 + (jj)])
#define IG(i, jj) (-gb[(i) * 10 + 4 + (jj)])

  float llr_[16];
  {
    const float4* lp4 = (const float4*)(llrp + (size_t)s2 * 16);
#pragma unroll
    for (int q = 0; q < 4; ++q) {
      float4 v = lp4[q];
      llr_[q * 4 + 0] = v.x; llr_[q * 4 + 1] = v.y;
      llr_[q * 4 + 2] = v.z; llr_[q * 4 + 3] = v.w;
    }
  }
  const float noval = nop[0];

  // priors -> soft symbols s_hat and error variances
  float shr[4], shi[4], ev[4], Kt[4];
#pragma unroll
  for (int k = 0; k < 4; ++k) {
    float p0b[4];
    float kt = 0.f;
#pragma unroll
    for (int b = 0; b < 4; ++b) {
      float v = llr_[k * 4 + b];
      p0b[b] = 1.f / (1.f + __expf(v));  // 0.5*(1-tanh(x/2)) == sigmoid(-x)
      float a = fabsf(v);
      kt += 0.5f * a + log1pf(__expf(-a));
    }
    Kt[k] = kt;
    float sr = 0.f, si = 0.f;
    float pcs[16];
#pragma unroll
    for (int p = 0; p < 16; ++p) {
      float pc = 1.f;
#pragma unroll
      for (int b = 0; b < 4; ++b) {
        int bit = (p >> (3 - b)) & 1;
        pc *= bit ? (1.f - p0b[b]) : p0b[b];
      }
      pcs[p] = pc;
      sr += ptsr(p) * pc;
      si += ptsi(p) * pc;
    }
    float e = 0.f;
#pragma unroll
    for (int p = 0; p < 16; ++p) {
      float dr = sr - ptsr(p), di = si - ptsi(p);
      e += fmaxf(dr * dr + di * di, EPS2F) * pcs[p];  // max(|d|,EPS)^2
    }
    shr[k] = sr; shi[k] = si; ev[k] = e;
  }

  // A = G*diag(ev) + no*I ; invert (complex 4x4 Gauss-Jordan, diag-dominant -> no pivot)
  float ar[4][4], ai[4][4], br[4][4], bi[4][4];
#pragma unroll
  for (int i = 0; i < 4; ++i)
#pragma unroll
    for (int jj = 0; jj < 4; ++jj) {
      ar[i][jj] = RG(i, jj) * ev[jj] + ((i == jj) ? noval : 0.f);
      ai[i][jj] = IG(i, jj) * ev[jj];
      br[i][jj] = (i == jj) ? 1.f : 0.f;
      bi[i][jj] = 0.f;
    }
#pragma unroll
  for (int p = 0; p < 4; ++p) {
    float dr = ar[p][p], di = ai[p][p];
    float iden = 1.f / (dr * dr + di * di);
    float qr = dr * iden, qi = -di * iden;
#pragma unroll
    for (int jj = 0; jj < 4; ++jj) {
      float xr = ar[p][jj], xi = ai[p][jj];
      ar[p][jj] = xr * qr - xi * qi; ai[p][jj] = xr * qi + xi * qr;
      float yr = br[p][jj], yi = bi[p][jj];
      br[p][jj] = yr * qr - yi * qi; bi[p][jj] = yr * qi + yi * qr;
    }
#pragma unroll
    for (int r = 0; r < 4; ++r) {
      if (r != p) {
        float fr = ar[r][p], fi = ai[r][p];
#pragma unroll
        for (int jj = 0; jj < 4; ++jj) {
          float pr_ = ar[p][jj], pi_ = ai[p][jj];
          ar[r][jj] -= fr * pr_ - fi * pi_;
          ai[r][jj] -= fr * pi_ + fi * pr_;
          float ur = br[p][jj], ui = bi[p][jj];
          br[r][jj] -= fr * ur - fi * ui;
          bi[r][jj] -= fr * ui + fi * ur;
        }
      }
    }
  }

  // mu, residual, z, rho
  float mu[4], rho[4], zr[4], zi[4], rr_[4], ri_[4];
#pragma unroll
  for (int i = 0; i < 4; ++i) {
    float r_ = gb[i * 10 + 8];    // Re y_mf
    float q_ = -gb[i * 10 + 9];   // Im y_mf
#pragma unroll
    for (int jj = 0; jj < 4; ++jj) {
      r_ -= RG(i, jj) * shr[jj] - IG(i, jj) * shi[jj];
      q_ -= RG(i, jj) * shi[jj] + IG(i, jj) * shr[jj];
    }
    rr_[i] = r_; ri_[i] = q_;
  }
#pragma unroll
  for (int k = 0; k < 4; ++k) {
    float mm = 0.f;
#pragma unroll
    for (int jj = 0; jj < 4; ++jj)
      mm += br[k][jj] * RG(jj, k) - bi[k][jj] * IG(jj, k);  // Re((A^-1 G)_kk)
    mu[k] = fmaxf(mm, EPSF);
    float ur = 0.f, ui = 0.f;
#pragma unroll
    for (int jj = 0; jj < 4; ++jj) {
      ur += br[k][jj] * rr_[jj] - bi[k][jj] * ri_[jj];
      ui += br[k][jj] * ri_[jj] + bi[k][jj] * rr_[jj];
    }
    zr[k] = ur / mu[k] + shr[k];
    zi[k] = ui / mu[k] + shi[k];
    rho[k] = mu[k] / fmaxf(1.f - ev[k] * mu[k], EPSF);
  }

  // extrinsic LLRs: per (k,bit) two-group logsumexp over 8 points each
  float4* op = (float4*)(outp + (size_t)s2 * 16);
#pragma unroll
  for (int k = 0; k < 4; ++k) {
    float e_[16];
#pragma unroll
    for (int p = 0; p < 16; ++p) {
      float dr = zr[k] - ptsr(p), di = zi[k] - ptsi(p);
      float d2 = fmaxf(dr * dr + di * di, EPS2F);
      float lpc = -Kt[k];
#pragma unroll
      for (int b = 0; b < 4; ++b) {
        int bit = (p >> (3 - b)) & 1;
        lpc += (bit ? 0.5f : -0.5f) * llr_[k * 4 + b];
      }
      e_[p] = -d2 * rho[k] + lpc;
    }
    float ov[4];
#pragma unroll
    for (int b = 0; b < 4; ++b) {
      float m0 = -3.0e38f, m1 = -3.0e38f;
#pragma unroll
      for (int p = 0; p < 16; ++p) {
        if (((p >> (3 - b)) & 1) != 0) m1 = fmaxf(m1, e_[p]);
        else                           m0 = fmaxf(m0, e_[p]);
      }
      float s0 = 0.f, s1 = 0.f;
#pragma unroll
      for (int p = 0; p < 16; ++p) {
        if (((p >> (3 - b)) & 1) != 0) s1 += __expf(e_[p] - m1);
        else                           s0 += __expf(e_[p] - m0);
      }
      ov[b] = (m1 + __logf(s1)) - (m0 + __logf(s0));
    }
    op[k] = make_float4(ov[0], ov[1], ov[2], ov[3]);
  }
#undef RG
#undef IG
}

extern "C" void kernel_launch(void* const* d_in, const int* in_sizes, int n_in,
                              void* d_out, int out_size, void* d_ws, size_t ws_size,
                              hipStream_t stream) {
  (void)n_in; (void)d_ws; (void)ws_size; (void)out_size;
  const float* y    = (const float*)d_in[0];
  const float* h    = (const float*)d_in[1];
  const float* llra = (const float*)d_in[2];
  const float* no   = (const float*)d_in[3];
  float* out = (float*)d_out;
  const int nsub = in_sizes[0] / 32;  // y = nsub * M * 2
  const int blocks = (nsub + 255) / 256;
  siso_mmse_pic_kernel<<<blocks, 256, 0, stream>>>(y, h, llra, no, out, nsub);
}